// DiagKernel_22170621182066
// MI455X (gfx1250) — compile-verified
//
#include <hip/hip_runtime.h>

// out[b, n] = x[b, n] * kernel[n, n]   (B=8192 rows, N=4096 cols, fp32)
//
// Memory-bound streaming kernel: 128 MiB in + 128 MiB out @ 23.3 TB/s ~= 11.5 us.
// WMMA is deliberately NOT used (50x memory-bound; matrix ops cannot help).
// CDNA5 paths used: global_load_async_to_lds (strided diagonal gather into LDS,
// ASYNCcnt + s_wait_asynccnt), B128 vector loads/stores, non-temporal hints.

typedef __attribute__((ext_vector_type(4))) float v4f;

// Address-space-qualified pointer types for the async-to-LDS builtin
// (probe: first param is a non-const AS1 int*, i.e. `__device__ int *`).
typedef __attribute__((address_space(1))) int g_int;   // global
typedef __attribute__((address_space(3))) int l_int;   // LDS

#define NCOLS          4096
#define BLOCK          256
#define ROWS_PER_BLOCK 8

__global__ __launch_bounds__(BLOCK) void diag_scale_kernel(
    const float* __restrict__ x,
    const float* __restrict__ kmat,
    float* __restrict__ out,
    int nrows)
{
    __shared__ float sdiag[NCOLS];   // 16 KB of 320 KB/WGP — high occupancy

    // ---- Stage the (strided) diagonal into LDS ----
    // kernel[n, n] lives at flat offset n*(NCOLS+1): a stride-4097 gather.
    // Each lane issues its own per-lane global address -> per-lane LDS address,
    // which is exactly the GLOBAL_LOAD_ASYNC_TO_LDS addressing model.
#if defined(__gfx1250__) && \
    __has_builtin(__builtin_amdgcn_global_load_async_to_lds_b32) && \
    __has_builtin(__builtin_amdgcn_s_wait_asynccnt)
    #pragma unroll
    for (int i = 0; i < NCOLS / BLOCK; ++i) {           // 16 async b32 gathers/lane
        const int col = i * BLOCK + threadIdx.x;
        __builtin_amdgcn_global_load_async_to_lds_b32(
            (g_int*)(kmat + (size_t)col * (NCOLS + 1)),
            (l_int*)(&sdiag[col]),
            /*imm offset=*/0, /*cpol=*/0);
    }
    __builtin_amdgcn_s_wait_asynccnt(0);                // s_wait_asynccnt 0
    __syncthreads();
#else
    // Fallback: plain strided loads + LDS stores (DScnt path)
    #pragma unroll
    for (int i = 0; i < NCOLS / BLOCK; ++i) {
        const int col = i * BLOCK + threadIdx.x;
        sdiag[col] = kmat[(size_t)col * (NCOLS + 1)];
    }
    __syncthreads();
#endif

    // ---- Stream rows: B128 loads, scale, B128 stores, all non-temporal ----
    const v4f* __restrict__ x4  = (const v4f*)x;
    v4f*       __restrict__ o4  = (v4f*)out;
    const v4f*              sd4 = (const v4f*)sdiag;

    const int row0 = blockIdx.x * ROWS_PER_BLOCK;
    #pragma unroll
    for (int r = 0; r < ROWS_PER_BLOCK; ++r) {
        const int row = row0 + r;
        if (row >= nrows) break;
        const size_t base = (size_t)row * (NCOLS / 4);
        #pragma unroll
        for (int i = 0; i < (NCOLS / 4) / BLOCK; ++i) { // 4 x float4 per lane per row
            const int c = i * BLOCK + threadIdx.x;
            v4f v  = __builtin_nontemporal_load(&x4[base + c]);  // read-once: NT
            v4f dv = sd4[c];                                     // ds_load_b128
            v *= dv;
            __builtin_nontemporal_store(v, &o4[base + c]);       // write-once: NT
        }
    }
}

extern "C" void kernel_launch(void* const* d_in, const int* in_sizes, int n_in,
                              void* d_out, int out_size, void* d_ws, size_t ws_size,
                              hipStream_t stream) {
    (void)n_in; (void)out_size; (void)d_ws; (void)ws_size;

    const float* x    = (const float*)d_in[0];   // [B, N] fp32
    const float* kmat = (const float*)d_in[1];   // [N, N] fp32
    float*       out  = (float*)d_out;           // [B, N] fp32

    const int nrows = in_sizes[0] / NCOLS;       // B = 8192
    const int grid  = (nrows + ROWS_PER_BLOCK - 1) / ROWS_PER_BLOCK;  // 1024 blocks

    diag_scale_kernel<<<grid, BLOCK, 0, stream>>>(x, kmat, out, nrows);
}